// Vandermonde_72473278153221
// MI455X (gfx1250) — compile-verified
//
#include <hip/hip_runtime.h>
#include <hip/hip_bf16.h>

typedef __attribute__((ext_vector_type(2))) float v2f;
typedef __attribute__((ext_vector_type(8))) float v8f;

#define PP   512      // d_state (K)
#define HH   512      // d_output (M)
#define LL   32768    // kernel length (N)
#define NT   32       // N columns per block (two 16-col WMMA subtiles)
#define KSTR 514      // padded LDS k-stride (float units) -> conflict-free b64 loads

// ---------------------------------------------------------------------------
// Kernel 1: CB[h,p] = C[h,p] * B[p,h]  (complex), stored pre-swizzled into
// WMMA A-fragment order:
//   cb[((t*128 + c)*32 + lane)*2 + j]  with  h = t*16 + (lane&15),
//   p = c*4 + 2*(lane>>4) + j
// so the GEMM loop's A-operand load is one coalesced global_load_b64/lane.
// ---------------------------------------------------------------------------
__global__ __launch_bounds__(256) void Vand_cb_precompute(
    const float* __restrict__ B, const float* __restrict__ C,
    float* __restrict__ cbr, float* __restrict__ cbi) {
  int tid = blockIdx.x * 256 + threadIdx.x;   // 0 .. H*P-1 == 262143
  int t    = tid >> 13;                       // 16-row M tile (32)
  int rem  = tid & 8191;
  int c    = rem >> 6;                        // K chunk of 4 (128)
  int rem2 = rem & 63;
  int l    = rem2 >> 1;                       // lane (32)
  int j    = rem2 & 1;                        // element within K-pair
  int h = t * 16 + (l & 15);
  int p = c * 4 + 2 * (l >> 4) + j;
  float cr = C[((size_t)h * PP + p) * 2 + 0];
  float ci = C[((size_t)h * PP + p) * 2 + 1];
  float br = B[((size_t)p * HH + h) * 2 + 0];
  float bi = B[((size_t)p * HH + h) * 2 + 1];
  cbr[tid] = cr * br - ci * bi;
  cbi[tid] = cr * bi + ci * br;
}

// ---------------------------------------------------------------------------
// Kernel 2: out[h, l] = sum_p CB[h,p] * A[p]^l   (complex GEMM via WMMA f32)
// Grid: (L/NT, H/128).  Block: 256 threads = 8 waves; wave w owns rows
// h0 = by*128 + w*16 and both 16-column subtiles of this NT-column tile.
// Phase 1: build V tile (512 x NT complex) in LDS from A (binary exponent
//          for A^l0, then incremental multiplies).
// Phase 2: K-loop of 128 chunks, 6x V_WMMA_F32_16X16X4_F32 per chunk
//          (Karatsuba: rr, ii, (r+i)(r+i) accumulators for 2 subtiles).
// ---------------------------------------------------------------------------
__global__ __launch_bounds__(256) void Vand_gemm_wmma(
    const float* __restrict__ A,
    const float* __restrict__ cbr, const float* __restrict__ cbi,
    float* __restrict__ out) {
  __shared__ float sVr[NT * KSTR];
  __shared__ float sVi[NT * KSTR];

  const int tid = threadIdx.x;
  const int l0  = blockIdx.x * NT;

  // ---- Phase 1: V[p, l0+j] for p in [0,512), j in [0,NT) ----
  #pragma unroll
  for (int pi = 0; pi < 2; ++pi) {
    int p = tid + pi * 256;
    float ax = A[2 * p + 0];
    float ay = A[2 * p + 1];
    // base = A[p]^l0 by binary exponentiation (uniform exponent)
    float rx = 1.0f, ry = 0.0f;
    float sx = ax, sy = ay;
    unsigned e = (unsigned)l0;
    while (e) {
      if (e & 1u) { float tx = rx * sx - ry * sy; ry = rx * sy + ry * sx; rx = tx; }
      float tx = sx * sx - sy * sy; sy = 2.0f * sx * sy; sx = tx;
      e >>= 1;
    }
    #pragma unroll
    for (int j = 0; j < NT; ++j) {
      sVr[j * KSTR + p] = rx;
      sVi[j * KSTR + p] = ry;
      float tx = rx * ax - ry * ay; ry = rx * ay + ry * ax; rx = tx;
    }
  }
  __syncthreads();

  // ---- Phase 2: WMMA K-loop ----
  const int wave = tid >> 5;
  const int lane = tid & 31;
  const int m    = lane & 15;        // M row within fragment / N col for B
  const int kh   = lane >> 4;        // K-pair selector
  const int h0   = blockIdx.y * 128 + wave * 16;
  const int t    = h0 >> 4;          // swizzled-CB tile index

  const float* __restrict__ cbrp = cbr + (size_t)t * 128 * 64;
  const float* __restrict__ cbip = cbi + (size_t)t * 128 * 64;

  v8f rr0 = {}, ii0 = {}, ss0 = {};
  v8f rr1 = {}, ii1 = {}, ss1 = {};

  #pragma unroll 4
  for (int c = 0; c < 128; ++c) {
    // A-fragments: coalesced 8B/lane stream from L2-resident swizzled CB
    v2f ar = *(const v2f*)(cbrp + c * 64 + lane * 2);
    v2f ai = *(const v2f*)(cbip + c * 64 + lane * 2);
    v2f as = ar + ai;
    // B-fragments from LDS (padded stride -> conflict-free ds_load_b64)
    int kk = c * 4 + kh * 2;
    v2f br0 = *(const v2f*)&sVr[(m     ) * KSTR + kk];
    v2f bi0 = *(const v2f*)&sVi[(m     ) * KSTR + kk];
    v2f br1 = *(const v2f*)&sVr[(m + 16) * KSTR + kk];
    v2f bi1 = *(const v2f*)&sVi[(m + 16) * KSTR + kk];
    v2f bs0 = br0 + bi0;
    v2f bs1 = br1 + bi1;
    // 3-multiply complex GEMM (Karatsuba), two N-subtiles
    rr0 = __builtin_amdgcn_wmma_f32_16x16x4_f32(false, ar, false, br0, (short)0, rr0, false, false);
    ii0 = __builtin_amdgcn_wmma_f32_16x16x4_f32(false, ai, false, bi0, (short)0, ii0, false, false);
    ss0 = __builtin_amdgcn_wmma_f32_16x16x4_f32(false, as, false, bs0, (short)0, ss0, false, false);
    rr1 = __builtin_amdgcn_wmma_f32_16x16x4_f32(false, ar, false, br1, (short)0, rr1, false, false);
    ii1 = __builtin_amdgcn_wmma_f32_16x16x4_f32(false, ai, false, bi1, (short)0, ii1, false, false);
    ss1 = __builtin_amdgcn_wmma_f32_16x16x4_f32(false, as, false, bs1, (short)0, ss1, false, false);
  }

  // ---- Combine + store: D layout is M = r + 8*(lane>>4), N = lane&15 ----
  const int mrow = (lane >> 4) * 8;
  const int n    = lane & 15;
  #pragma unroll
  for (int r = 0; r < 8; ++r) {
    int h = h0 + mrow + r;
    size_t off = ((size_t)h * LL + (size_t)(l0 + n)) * 2;
    float kr0 = rr0[r] - ii0[r];
    float ki0 = ss0[r] - rr0[r] - ii0[r];
    *(v2f*)(out + off) = (v2f){kr0, ki0};
    float kr1 = rr1[r] - ii1[r];
    float ki1 = ss1[r] - rr1[r] - ii1[r];
    *(v2f*)(out + off + 32) = (v2f){kr1, ki1};   // second subtile: l += 16
  }
}

// ---------------------------------------------------------------------------
extern "C" void kernel_launch(void* const* d_in, const int* in_sizes, int n_in,
                              void* d_out, int out_size, void* d_ws, size_t ws_size,
                              hipStream_t stream) {
  (void)in_sizes; (void)n_in; (void)out_size; (void)ws_size;
  const float* A = (const float*)d_in[0];   // (P, 2)
  const float* B = (const float*)d_in[1];   // (P, H, 2)
  const float* C = (const float*)d_in[2];   // (H, P, 2)
  // d_in[3] = powers (unused: powers[l] == l)
  float* out = (float*)d_out;               // (H, L, 2) complex64 viewed as f32
  float* cbr = (float*)d_ws;                // H*P floats (swizzled real part)
  float* cbi = cbr + (size_t)HH * PP;       // H*P floats (swizzled imag part)

  Vand_cb_precompute<<<(HH * PP) / 256, 256, 0, stream>>>(B, C, cbr, cbi);

  dim3 grid(LL / NT, HH / 128);
  Vand_gemm_wmma<<<grid, 256, 0, stream>>>(A, cbr, cbi, out);
}